// GNNLayer_39857296507014
// MI455X (gfx1250) — compile-verified
//
#include <hip/hip_runtime.h>
#include <math.h>

// Problem constants (fixed by the reference)
#define N_NODE   50000
#define N_EDGE   800000
#define IN_DIM   128
#define OUT_DIM  128
#define ATTN_DIM 64
#define N_REL    200
#define N_Q      64
#define N_EMB    (2 * N_REL + 1)   // 401

typedef __attribute__((ext_vector_type(2))) float v2f;
typedef __attribute__((ext_vector_type(8))) float v8f;

// ---------------------------------------------------------------------------
// Kernel 1: tiny projections
//   rproj[r, c] = sum_k rela_embed[r,k] * Wr[k,c]                 (401 x 64)
//   qproj[q, c] = sum_k rela_embed[q_rel[q],k]*Wqr_W[k,c] + b[c]  (64 x 64)
// ---------------------------------------------------------------------------
__global__ void small_proj_kernel(const float* __restrict__ rela,
                                  const float* __restrict__ Wr,
                                  const float* __restrict__ WqrW,
                                  const float* __restrict__ Wqrb,
                                  const int*   __restrict__ q_rel,
                                  float* __restrict__ rproj,
                                  float* __restrict__ qproj) {
    int idx = blockIdx.x * blockDim.x + threadIdx.x;
    const int total = (N_EMB + N_Q) * ATTN_DIM;
    if (idx >= total) return;
    int col = idx & (ATTN_DIM - 1);
    int row = idx >> 6;
    if (row < N_EMB) {
        const float* h = rela + (size_t)row * IN_DIM;
        float s = 0.f;
        #pragma unroll 8
        for (int k = 0; k < IN_DIM; ++k) s = fmaf(h[k], Wr[k * ATTN_DIM + col], s);
        rproj[row * ATTN_DIM + col] = s;
    } else {
        int q = row - N_EMB;
        const float* h = rela + (size_t)q_rel[q] * IN_DIM;
        float s = Wqrb[col];
        #pragma unroll 8
        for (int k = 0; k < IN_DIM; ++k) s = fmaf(h[k], WqrW[k * ATTN_DIM + col], s);
        qproj[q * ATTN_DIM + col] = s;
    }
}

// ---------------------------------------------------------------------------
// WMMA f32 GEMM:  C[M x N] = A[M x K] @ B[K x N], row-major, K = 128,
// M multiple of 16, N in {64, 128}. Block = 256 threads = 8 waves.
// Each wave computes one 16x64 C tile using V_WMMA_F32_16X16X4_F32
// (4 accumulators, 32 K-steps).
//
// B is staged once per block into LDS in K-PAIR-INTERLEAVED layout:
//   Bs2[k/2][n] = { B[k][n], B[k+1][n] }   (8-byte unit)
// so each lane's B fragment (rows k,k+1 at one column) is a single aligned
// ds_load_b64 straight into an even VGPR pair — no packing moves.
//
// ISA lane striping (wave32):
//   A 16x4:  lanes 0-15 -> M=lane,  VGPR0=K0, VGPR1=K1
//            lanes16-31 -> M=l-16,  VGPR0=K2, VGPR1=K3
//   B 4x16:  lanes 0-15 -> N=lane,  VGPR0=K0 row, VGPR1=K1 row
//            lanes16-31 -> N=l-16,  VGPR0=K2 row, VGPR1=K3 row
//   C 16x16: VGPR r: lanes 0-15 -> (M=r,   N=lane)
//                    lanes16-31 -> (M=r+8, N=lane-16)
// ---------------------------------------------------------------------------
template <int N>
__global__ void gemm_wmma_f32_kernel(const float* __restrict__ A,
                                     const float* __restrict__ B,
                                     float* __restrict__ C,
                                     int M) {
    constexpr int K = 128;
    __shared__ v2f Bs2[(K / 2) * N];   // pair-interleaved B

    // Stage B (K x N) into LDS as {B[2k][n], B[2k+1][n]} pairs.
    for (int i = threadIdx.x; i < (K / 2) * N; i += blockDim.x) {
        const int kp = i / N;          // K pair index
        const int n  = i - kp * N;
        v2f p;
        p.x = B[(2 * kp)     * N + n];
        p.y = B[(2 * kp + 1) * N + n];
        Bs2[i] = p;
    }
    __syncthreads();

    const int lane = threadIdx.x & 31;
    const int waveInBlock = threadIdx.x >> 5;
    constexpr int colTiles = N / 64;               // 64-wide wave tiles per row strip
    const long tileId = (long)blockIdx.x * 8 + waveInBlock;
    const int  mTile  = (int)(tileId / colTiles);
    const int  nTile  = (int)(tileId % colTiles);
    if (mTile * 16 >= M) return;                    // uniform per wave -> EXEC stays all-1s

    const int mrow  = mTile * 16 + (lane & 15);     // A row this lane feeds
    const int sel   = lane >> 4;                    // 0: K0/K1 rows, 1: K2/K3 rows
    const int koff  = sel * 2;
    const int ncol0 = nTile * 64;
    const int nlane = ncol0 + (lane & 15);          // base B column this lane feeds

    v8f acc0 = {}, acc1 = {}, acc2 = {}, acc3 = {};

    const float* __restrict__ arow = A + (size_t)mrow * K;
    #pragma unroll 4
    for (int k0 = 0; k0 < K; k0 += 4) {
        v2f a;
        a.x = arow[k0 + koff];
        a.y = arow[k0 + koff + 1];

        // pair row this lane needs: k0/2 (+1 for lanes 16-31)
        const v2f* __restrict__ brow = &Bs2[(k0 / 2 + sel) * N + nlane];
        acc0 = __builtin_amdgcn_wmma_f32_16x16x4_f32(false, a, false, brow[ 0], (short)0, acc0, false, false);
        acc1 = __builtin_amdgcn_wmma_f32_16x16x4_f32(false, a, false, brow[16], (short)0, acc1, false, false);
        acc2 = __builtin_amdgcn_wmma_f32_16x16x4_f32(false, a, false, brow[32], (short)0, acc2, false, false);
        acc3 = __builtin_amdgcn_wmma_f32_16x16x4_f32(false, a, false, brow[48], (short)0, acc3, false, false);
    }

    // Write back per the C/D 16x16 layout (no acc array -> no extra moves).
    const int cmBase = mTile * 16 + ((lane < 16) ? 0 : 8);
    const int cn = lane & 15;
    float* __restrict__ c0 = C + (size_t)cmBase * N + ncol0 + cn;
    #pragma unroll
    for (int r = 0; r < 8; ++r) {
        float* cr = c0 + (size_t)r * N;
        cr[ 0] = acc0[r];
        cr[16] = acc1[r];
        cr[32] = acc2[r];
        cr[48] = acc3[r];
    }
}

// ---------------------------------------------------------------------------
// Edge kernel: one edge per wave iteration (wave32).
//   pre[j] = hproj[sub,j] + rproj[rel,j] + qproj[r_idx,j]     (j = 0..63)
//   alpha  = sigmoid( relu(pre) . wa_W + wa_b )
//   atomicAdd(agg[obj, d], alpha * hidden[sub,d] * rela[rel,d])  (d = 0..127)
// All gather tables are L2-resident (hidden 25.6MB, hproj 12.8MB, agg 25.6MB
// inside the 192MB L2), so the scatter atomics stay on-chip.
// ---------------------------------------------------------------------------
__global__ void edge_kernel(const int*   __restrict__ edges,
                            const int*   __restrict__ r_idx,
                            const float* __restrict__ hidden,
                            const float* __restrict__ rela,
                            const float* __restrict__ hproj,
                            const float* __restrict__ rproj,
                            const float* __restrict__ qproj,
                            const float* __restrict__ waW,
                            const float* __restrict__ wab,
                            float* __restrict__ agg) {
    const int lane   = threadIdx.x & 31;
    const int wave   = (blockIdx.x * blockDim.x + threadIdx.x) >> 5;
    const int nwaves = (gridDim.x * blockDim.x) >> 5;

    const float bias = wab[0];
    const v2f wa2 = ((const v2f*)waW)[lane];        // wa_W[2*lane], wa_W[2*lane+1]

    for (int e = wave; e < N_EDGE; e += nwaves) {
        const int sub = edges[e * 3 + 0];
        const int rel = edges[e * 3 + 1];
        const int obj = edges[e * 3 + 2];
        const int q   = r_idx[e];

        // 64-dim attention pre-activation, 2 dims per lane (float2 loads).
        v2f hp = ((const v2f*)(hproj + (size_t)sub * ATTN_DIM))[lane];
        v2f rp = ((const v2f*)(rproj + (size_t)rel * ATTN_DIM))[lane];
        v2f qp = ((const v2f*)(qproj + (size_t)q   * ATTN_DIM))[lane];
        float p0 = fmaxf(hp.x + rp.x + qp.x, 0.f);
        float p1 = fmaxf(hp.y + rp.y + qp.y, 0.f);
        float s  = fmaf(p0, wa2.x, p1 * wa2.y);

        // wave32 all-reduce
        #pragma unroll
        for (int off = 16; off > 0; off >>= 1) s += __shfl_xor(s, off, 32);

        const float alpha = 1.0f / (1.0f + __expf(-(s + bias)));

        // 128-dim message scatter: 4 coalesced f32 atomics per lane.
        const float* hs = hidden + (size_t)sub * IN_DIM;
        const float* hr = rela   + (size_t)rel * IN_DIM;
        float* dst = agg + (size_t)obj * IN_DIM;
        #pragma unroll
        for (int d4 = 0; d4 < 4; ++d4) {
            const int d = lane + 32 * d4;
            atomicAdd(&dst[d], alpha * hs[d] * hr[d]);
        }
    }
}

// ---------------------------------------------------------------------------
// Launch
// ---------------------------------------------------------------------------
extern "C" void kernel_launch(void* const* d_in, const int* in_sizes, int n_in,
                              void* d_out, int out_size, void* d_ws, size_t ws_size,
                              hipStream_t stream) {
    // Inputs (setup_inputs order)
    const int*   q_rel  = (const int*)  d_in[1];
    const int*   r_idx  = (const int*)  d_in[2];
    const float* hidden = (const float*)d_in[3];
    const int*   edges  = (const int*)  d_in[4];
    const float* rela   = (const float*)d_in[6];
    const float* Ws     = (const float*)d_in[7];
    const float* Wr     = (const float*)d_in[8];
    const float* WqrW   = (const float*)d_in[9];
    const float* Wqrb   = (const float*)d_in[10];
    const float* waW    = (const float*)d_in[11];
    const float* wab    = (const float*)d_in[12];
    const float* Wh     = (const float*)d_in[13];
    float* out = (float*)d_out;

    // Workspace layout (256B aligned slices)
    char* ws = (char*)d_ws;
    float* agg   = (float*)(ws);                                     // 50000*128 f32 = 25.6 MB
    float* hproj = (float*)(ws + (size_t)N_NODE * IN_DIM * 4);       // 50000*64  f32 = 12.8 MB
    float* rproj = (float*)(ws + (size_t)N_NODE * IN_DIM * 4
                               + (size_t)N_NODE * ATTN_DIM * 4);     // 401*64 f32
    float* qproj = (float*)((char*)rproj + (((size_t)N_EMB * ATTN_DIM * 4 + 255ull) & ~255ull));

    // 1) zero the segment-sum accumulator (graph-capturable memset node)
    hipMemsetAsync(agg, 0, (size_t)N_NODE * IN_DIM * sizeof(float), stream);

    // 2) tiny projections rproj / qproj (bias folded into qproj)
    {
        const int total = (N_EMB + N_Q) * ATTN_DIM;
        small_proj_kernel<<<(total + 255) / 256, 256, 0, stream>>>(
            rela, Wr, WqrW, Wqrb, q_rel, rproj, qproj);
    }

    // 3) hproj = hidden @ Ws   (50000x128 @ 128x64) via WMMA f32
    {
        const int mTiles = N_NODE / 16;            // 3125, exact
        const int waveTiles = mTiles * (ATTN_DIM / 64);
        gemm_wmma_f32_kernel<ATTN_DIM><<<(waveTiles + 7) / 8, 256, 0, stream>>>(
            hidden, Ws, hproj, N_NODE);
    }

    // 4) edge pass: attention gate + message scatter
    edge_kernel<<<4096, 256, 0, stream>>>(edges, r_idx, hidden, rela,
                                          hproj, rproj, qproj, waW, wab, agg);

    // 5) out = agg @ Wh        (50000x128 @ 128x128) via WMMA f32
    {
        const int mTiles = N_NODE / 16;
        const int waveTiles = mTiles * (OUT_DIM / 64);  // 6250
        gemm_wmma_f32_kernel<OUT_DIM><<<(waveTiles + 7) / 8, 256, 0, stream>>>(
            agg, Wh, out, N_NODE);
    }
}